// SLAPInitEmbedding_71262097376026
// MI455X (gfx1250) — compile-verified
//
#include <hip/hip_runtime.h>
#include <hip/hip_bf16.h>

// out[b,n,:] = x*W[0,:] + y*W[1,:] + d*W[2,:] + bias,  d = L1(locs[b,n], locs[b,0])
// Computed as a 16x4 @ 4x16 f32 WMMA per output tile (K padded 3->4 with zeros),
// exact f32 math via V_WMMA_F32_16X16X4_F32. Store-bandwidth bound (~65.5 MB out).
typedef __attribute__((ext_vector_type(2))) float v2f;
typedef __attribute__((ext_vector_type(8))) float v8f;

#define NNODES 2000
#define EMBED 128
#define TILES_PER_BATCH 125   // 2000 / 16, exact -> tiles never straddle a batch

__global__ __launch_bounds__(128) void slap_init_embed_wmma(
    const float* __restrict__ locs,   // (B, 2000, 2) f32
    const float* __restrict__ W,      // (3, 128)     f32
    const float* __restrict__ bias,   // (128,)       f32
    float* __restrict__ out)          // (B, 2000, 128) f32
{
    const int lane  = threadIdx.x & 31;
    const int wave  = threadIdx.x >> 5;
    const int tileM = blockIdx.x * 4 + wave;          // 0..7999
    const int batch = tileM / TILES_PER_BATCH;
    const int row0  = (tileM % TILES_PER_BATCH) * 16; // row within batch

    const int  mlane = lane & 15;
    const bool hi    = lane >= 16;                    // lanes 16..31 -> K=2,3 half

    // ---- Preload B tiles (4x16 each) and bias for all 8 N-tiles ----
    // Per-lane base pointer select (no divergent loads): lo half reads W row 0,
    // hi half reads W row 2. Row 1 is loaded unconditionally, then masked to 0
    // in the hi half (K=3 zero padding) via cndmask.
    const float* __restrict__ wK0 = W + (hi ? 2 * EMBED : 0);
    const float* __restrict__ wK1 = W + EMBED;

    v2f   bm[8];
    float bv[8];
    #pragma unroll
    for (int nt = 0; nt < 8; ++nt) {
        const int n = nt * 16 + mlane;
        bm[nt].x = wK0[n];                   // K=0 (lo) / K=2 (hi)
        float w1 = wK1[n];
        bm[nt].y = hi ? 0.0f : w1;           // K=1 (lo) / K=3 = 0 (hi)
        bv[nt]   = bias[n];
    }

    // ---- Build A matrix (16x4): feats row m = (x, y, d, 0) ----
    const v2f* lb = (const v2f*)(locs + (size_t)batch * NNODES * 2);
    v2f depot = lb[0];
    v2f p     = lb[row0 + mlane];
    float x = p.x, y = p.y;
    float d = fabsf(x - depot.x) + fabsf(y - depot.y); // L1 depot distance

    v2f a;
    a.x = hi ? d    : x;   // VGPR0: K=0 (lanes 0-15) / K=2 (lanes 16-31)
    a.y = hi ? 0.0f : y;   // VGPR1: K=1 (lanes 0-15) / K=3 (lanes 16-31)

    float* orow = out + ((size_t)batch * NNODES + row0) * EMBED;
    const int mbase = hi ? 8 : 0;   // D layout: VGPR r -> row r (lo half) / r+8 (hi half)

    // ---- 8 WMMAs, each producing a finished 16x16 output tile ----
    #pragma unroll
    for (int nt = 0; nt < 8; ++nt) {
        const int n = nt * 16 + mlane;

        v8f c;
        #pragma unroll
        for (int r = 0; r < 8; ++r) c[r] = bv[nt];   // C = bias broadcast along M

        // (neg_a, A, neg_b, B, c_mod, C, reuse_a, reuse_b)
        c = __builtin_amdgcn_wmma_f32_16x16x4_f32(
                false, a, false, bm[nt], (short)0, c, false, false);

        #pragma unroll
        for (int r = 0; r < 8; ++r) {
            orow[(size_t)(mbase + r) * EMBED + n] = c[r];  // half-wave 64B runs
        }
    }
}

extern "C" void kernel_launch(void* const* d_in, const int* in_sizes, int n_in,
                              void* d_out, int out_size, void* d_ws, size_t ws_size,
                              hipStream_t stream) {
    const float* locs = (const float*)d_in[0];  // 64*2000*2
    const float* W    = (const float*)d_in[1];  // 3*128
    const float* bias = (const float*)d_in[2];  // 128
    float* out        = (float*)d_out;          // 64*2000*128

    // 8000 M-tiles of 16 rows; 4 waves (128 threads) per block.
    slap_init_embed_wmma<<<2000, 128, 0, stream>>>(locs, W, bias, out);
}